// Get_Abstract_Caps_Down_26010321945023
// MI455X (gfx1250) — compile-verified
//
#include <hip/hip_runtime.h>
#include <math.h>

// ---------------------------------------------------------------------------
// Fully fused capsule layer for MI455X (gfx1250, wave32).
//
// One block per (b, o) pair (256 blocks). Each block:
//   1. stages A = primary_caps[b,o] (256x32 f32) and the gathered/transposed
//      B = W[b,o] (32x512) into LDS,
//   2. runs the 256x32 @ 32x512 f32 GEMM with V_WMMA_F32_16X16X4_F32 in
//      16-row strips, parking each strip in LDS,
//   3. immediately performs agreement routing on the strip (one wave32 per
//      output position; lane == input-module m, IM == 32 == warpSize),
//      writing the final v vectors.
// No global intermediate: the 134 MB x tensor never leaves the WGP.
// ---------------------------------------------------------------------------

typedef __attribute__((ext_vector_type(2))) float v2f;
typedef __attribute__((ext_vector_type(8))) float v8f;

#define ALDS_STRIDE 36    // 256 x 32 A, padded: conflict-free frag reads, 16B rows
#define BLDS_STRIDE 516   // 32 x 512 B, padded
#define SLDS_STRIDE 524   // 16 x 512 output strip, padded
#define LDS_FLOATS  (256 * ALDS_STRIDE + 32 * BLDS_STRIDE + 16 * SLDS_STRIDE)

__device__ __forceinline__ float lane_sum(float v) {
#pragma unroll
    for (int off = 16; off > 0; off >>= 1) v += __shfl_xor(v, off, 32);
    return v;
}
__device__ __forceinline__ float lane_max(float v) {
#pragma unroll
    for (int off = 16; off > 0; off >>= 1) v = fmaxf(v, __shfl_xor(v, off, 32));
    return v;
}

__device__ __forceinline__ void weighted_squash(const float* __restrict__ xv,
                                                float c, float* __restrict__ sv)
{
    float t[16];
#pragma unroll
    for (int d = 0; d < 16; ++d) t[d] = xv[d] * c;
#pragma unroll
    for (int off = 16; off > 0; off >>= 1) {
#pragma unroll
        for (int d = 0; d < 16; ++d) t[d] += __shfl_xor(t[d], off, 32);
    }
    float sq = 0.f;
#pragma unroll
    for (int d = 0; d < 16; ++d) sq = fmaf(t[d], t[d], sq);
    float coef = (sq / (1.0f + sq)) / sqrtf(sq + 1e-7f);
#pragma unroll
    for (int d = 0; d < 16; ++d) sv[d] = coef * t[d];
}

__global__ __launch_bounds__(256)
void capsule_fused_wmma(const float* __restrict__ prim,   // [256][256*32]
                        const float* __restrict__ Wt,     // [256][4*32*16*8]
                        const float* __restrict__ bprior, // [32*256][32]
                        float* __restrict__ out)          // [256*256][16]
{
    extern __shared__ float smem[];
    float* Alds  = smem;                             // [256][ALDS_STRIDE]
    float* Blds  = Alds + 256 * ALDS_STRIDE;         // [32][BLDS_STRIDE]
    float* Strip = Blds + 32 * BLDS_STRIDE;          // [16][SLDS_STRIDE]

    const int bm   = blockIdx.x;        // b*32 + o
    const int o    = bm & 31;
    const int tid  = threadIdx.x;
    const int wave = tid >> 5;          // 8 waves of 32
    const int lane = tid & 31;

    const float* Ag = prim + (size_t)bm * 8192;      // 256x32 row-major (raw reshape)
    const float* Wg = Wt   + (size_t)bm * 16384;     // [yx=4][og=32][d=16][e=8]

    __builtin_prefetch(Wg, 0, 0);                    // global_prefetch_b8

    // ---- stage A (32KB straight copy, 16B-aligned LDS rows) ----
#pragma unroll
    for (int i = 0; i < 8; ++i) {
        int f4  = tid + i * 256;                     // float4 index 0..2047
        float4 v = ((const float4*)Ag)[f4];
        int f   = f4 << 2;
        int row = f >> 5;
        int kk  = f & 31;
        float* dst = &Alds[row * ALDS_STRIDE + kk];
        dst[0] = v.x; dst[1] = v.y; dst[2] = v.z; dst[3] = v.w;
    }
    // ---- stage/transpose full B: Blds[c][n] = W[yx][og][d][e], c=yx*8+e, n=og*16+d ----
#pragma unroll 4
    for (int i = 0; i < 64; ++i) {
        int lin = i * 256 + tid;                     // 0..16383
        int c   = lin >> 9;                          // 0..31
        int n   = lin & 511;
        int yx  = c >> 3, e = c & 7;
        int og  = n >> 4, d = n & 15;
        Blds[c * BLDS_STRIDE + n] = Wg[yx * 4096 + og * 128 + d * 8 + e];
    }
    __syncthreads();

    const int khalf = (lane >> 4) << 1;              // lanes 16-31 hold K+2,K+3

    for (int mtile = 0; mtile < 16; ++mtile) {
        // ---- GEMM strip: rows k = mtile*16..+15, all 512 cols; 4 N-tiles/wave ----
        const int arow = mtile * 16 + (lane & 15);
#pragma unroll
        for (int i = 0; i < 4; ++i) {
            int nt  = i * 8 + wave;                  // N-tile 0..31 (== input module m)
            int col = nt * 16 + (lane & 15);
            v8f acc = {};
#pragma unroll
            for (int kc = 0; kc < 32; kc += 4) {
                int kk = kc + khalf;
                v2f a, b;
                // A 16x4 f32 frag (ISA layout): lane<16 -> K=kc,kc+1; lane>=16 -> kc+2,kc+3
                a.x = Alds[arow * ALDS_STRIDE + kk];
                a.y = Alds[arow * ALDS_STRIDE + kk + 1];
                // B 4x16 f32 frag, column striped across lanes
                b.x = Blds[kk * BLDS_STRIDE + col];
                b.y = Blds[(kk + 1) * BLDS_STRIDE + col];
                acc = __builtin_amdgcn_wmma_f32_16x16x4_f32(
                          false, a, false, b, (short)0, acc, false, false);
            }
            // D layout: VGPR r -> row r + (lane>=16 ? 8 : 0), col lane&15 of the tile.
            // Park tile in the strip: Strip[kRowInStrip][n].
            int rbase = (lane >> 4) << 3;
#pragma unroll
            for (int r = 0; r < 8; ++r)
                Strip[(rbase + r) * SLDS_STRIDE + col] = acc[r];
        }
        __syncthreads();

        // ---- routing on the 16 strip rows (positions k = mtile*16 + rowk) ----
#pragma unroll
        for (int p = 0; p < 2; ++p) {
            int rowk = p * 8 + wave;                 // 0..15
            int k    = mtile * 16 + rowk;            // h*16+w

            const float* sp = &Strip[rowk * SLDS_STRIDE + lane * 16];
            float xv[16];
#pragma unroll
            for (int q = 0; q < 4; ++q) {
                float4 v = ((const float4*)sp)[q];
                xv[q * 4 + 0] = v.x; xv[q * 4 + 1] = v.y;
                xv[q * 4 + 2] = v.z; xv[q * 4 + 3] = v.w;
            }
            float bb = bprior[((size_t)o * 256 + k) * 32 + lane];

            float sv[16];
            {   // c = softmax(b); v = squash(sum_m c*x)
                float mx = lane_max(bb);
                float e  = __expf(bb - mx);
                float s  = lane_sum(e);
                weighted_squash(xv, e / s, sv);
            }
#pragma unroll
            for (int it = 0; it < 2; ++it) {         // NUM_ITERATIONS = 2
                float dot = 0.f;
#pragma unroll
                for (int d = 0; d < 16; ++d) dot = fmaf(xv[d], sv[d], dot);
                bb += dot;
                float mx = lane_max(bb);
                float e  = __expf(bb - mx);
                float s  = lane_sum(e);
                weighted_squash(xv, e / s, sv);
            }

            if (lane == 0) {
                float4* op = (float4*)(out + ((size_t)bm * 256 + k) * 16);
                op[0] = make_float4(sv[0],  sv[1],  sv[2],  sv[3]);
                op[1] = make_float4(sv[4],  sv[5],  sv[6],  sv[7]);
                op[2] = make_float4(sv[8],  sv[9],  sv[10], sv[11]);
                op[3] = make_float4(sv[12], sv[13], sv[14], sv[15]);
            }
        }
        __syncthreads();                             // Strip reused next mtile
    }
}

extern "C" void kernel_launch(void* const* d_in, const int* in_sizes, int n_in,
                              void* d_out, int out_size, void* d_ws, size_t ws_size,
                              hipStream_t stream)
{
    const float* prim = (const float*)d_in[0];   // primary_caps [8,32,32,32,8]
    const float* Wt   = (const float*)d_in[1];   // W [8,32,1,2,2,32,16,8]
    const float* bpr  = (const float*)d_in[2];   // b [1,32,16,16,32]
    float* out = (float*)d_out;                  // [8,32,16,16,16]

    const size_t lds_bytes = (size_t)LDS_FLOATS * sizeof(float);  // ~133 KB of 320 KB
    capsule_fused_wmma<<<256, 256, lds_bytes, stream>>>(prim, Wt, bpr, out);
}